// FrFTLayer_25091198943874
// MI455X (gfx1250) — compile-verified
//
#include <hip/hip_runtime.h>

typedef __attribute__((ext_vector_type(2))) float v2f;
typedef __attribute__((ext_vector_type(8))) float v8f;
typedef __attribute__((ext_vector_type(4))) unsigned int v4u;
typedef __attribute__((ext_vector_type(8))) int v8i;
typedef __attribute__((ext_vector_type(4))) int v4i;

#if defined(__AMDGCN__) && __has_builtin(__builtin_amdgcn_tensor_load_to_lds)
#define USE_TDM 1
#else
#define USE_TDM 0
#endif

namespace {
constexpr int    NSIG  = 4096;
constexpr int    BATCH = 256;
constexpr double PI_D  = 3.14159265358979323846264338327950288;

// LDS tile geometry: 16 rows x 64 k, padded row stride 68 floats (16B aligned,
// bank-conflict-free for the 16-lane column reads of the A fragments).
constexpr int TROWS = 16, TCOLS = 64, TSTRIDE = 68;
constexpr int TILEF = TROWS * TSTRIDE;   // floats per staged plane (1088)

// Per-workgroup output width (8 waves x 2 tiles x 16 cols)
constexpr int WGCOLS = 256;

// LDS window sizes for the reversed Toeplitz tables
constexpr int WWIN = 4352;               // GEMM1: 4351 used
constexpr int KWIN = 8704;               // GEMM2: 8702 used (per plane)

// float-element offsets into workspace (d_ws)
constexpr size_t OFF_WTAB    = 64;                              // 2*NSIG
constexpr size_t OFF_CHRP_RE = OFF_WTAB    + 2 * NSIG;          // 16384
constexpr size_t OFF_CHRP_IM = OFF_CHRP_RE + 16384;             // 16384
constexpr size_t OFF_KERN_RE = OFF_CHRP_IM + 16384;             // 32768
constexpr size_t OFF_KERN_IM = OFF_KERN_RE + 32768;             // 32768
constexpr size_t OFF_FC_RE   = OFF_KERN_IM + 32768;             // BATCH*NSIG
constexpr size_t OFF_FC_IM   = OFF_FC_RE + (size_t)BATCH * NSIG;
constexpr size_t OFF_UP_RE   = OFF_FC_IM + (size_t)BATCH * NSIG;
constexpr size_t OFF_UP_IM   = OFF_UP_RE + (size_t)BATCH * NSIG;
constexpr size_t OFF_G_RE    = OFF_UP_IM + (size_t)BATCH * NSIG;
constexpr size_t OFF_G_IM    = OFF_G_RE + (size_t)BATCH * 2 * NSIG;
} // namespace

// fp32 WMMA: D(16x16) = A(16x4) * B(4x16) + C.
__device__ __forceinline__ v8f wmma4(v2f a, v2f b, v8f c) {
  return __builtin_amdgcn_wmma_f32_16x16x4_f32(false, a, false, b, (short)0, c,
                                               false, false);
}

#if USE_TDM
// Tensor Data Mover: DMA a 16x64 fp32 tile (global row stride = strideElems)
// into LDS at byte offset ldsOff, padding +4 DWORDs after every 64 DWORDs so
// LDS rows land at stride 68 floats.  D# packed per CDNA5 ISA section 8.3-8.5.
__device__ __forceinline__ void tdm_tile16x64(const float* gsrc, unsigned ldsOff,
                                              unsigned strideElems) {
  unsigned long long ga = (unsigned long long)(uintptr_t)gsrc;
  v4u g0;
  g0.x = 1u;                                   // count=1, user descriptor
  g0.y = ldsOff;                               // lds_addr (bytes)
  g0.z = (unsigned)ga;                         // global_addr[31:0]
  g0.w = ((unsigned)(ga >> 32) & 0x01FFFFFFu) | (2u << 30);  // addr[56:32]|type=2
  const unsigned td0 = 1u << 30, td1 = 1u << 30;             // huge tensor dims
  v8i g1;
  g1[0] = (int)((2u << 16) | (1u << 20) | (5u << 22) | (3u << 25));
          // data_size=4B | pad_enable | pad_interval: 64 DW | pad_amount: 4 DW
  g1[1] = (int)((td0 & 0xFFFFu) << 16);                      // tensor_dim0 lo
  g1[2] = (int)((td0 >> 16) | ((td1 & 0xFFFFu) << 16));      // dim0 hi | dim1 lo
  g1[3] = (int)((td1 >> 16) | ((unsigned)TCOLS << 16));      // dim1 hi | tile_dim0
  g1[4] = (int)((unsigned)TROWS);                            // tile_dim1 | tile_dim2=0
  g1[5] = (int)strideElems;                                  // tensor_dim0_stride lo
  g1[6] = 0;                                                 // stride hi | dim1_stride lo
  g1[7] = 0;
  v4i z4 = {0, 0, 0, 0};
#if __clang_major__ >= 23
  v8i z8 = {0, 0, 0, 0, 0, 0, 0, 0};
  __builtin_amdgcn_tensor_load_to_lds(g0, g1, z4, z4, z8, 0);
#else
  __builtin_amdgcn_tensor_load_to_lds(g0, g1, z4, z4, 0);
#endif
}
#endif

// Fallback cooperative staging: 256 threads move a 16x64 tile (one float4 each
// per plane) into the padded LDS layout.
__device__ __forceinline__ void stage_load(const float* g, size_t strideElems,
                                           int kb, int tid, float4* v) {
  const int rw = tid >> 4, q = (tid & 15) * 4;
  *v = *(const float4*)(g + (size_t)rw * strideElems + kb + q);
}
__device__ __forceinline__ void stage_store(float* s, int tid, const float4* v) {
  const int rw = tid >> 4, q = (tid & 15) * 4;
  *(float4*)(s + rw * TSTRIDE + q) = *v;
}

// ---------------------------------------------------------------------------
// K0: order -> scalars (double precision).
// ---------------------------------------------------------------------------
__global__ void k_setup(const float* __restrict__ order_p, double* __restrict__ P) {
  if (threadIdx.x == 0 && blockIdx.x == 0) {
    double a = fmod((double)order_p[0], 4.0);
    if (a < 0.0) a += 4.0;
    double c1 = (a > 2.0) ? 1.0 : 0.0; if (c1 != 0.0) a -= 2.0;
    double c2 = (a > 1.5) ? 1.0 : 0.0; if (c2 != 0.0) a -= 1.0;
    double c3 = (a < 0.5) ? 1.0 : 0.0; if (c3 != 0.0) a += 1.0;
    double alpha = a * PI_D * 0.5;
    double tana2 = tan(alpha * 0.5);
    double sina  = sin(alpha);
    double Cchrp = -(PI_D / (double)NSIG) * (tana2 * 0.25);
    double c     = PI_D / (double)NSIG / sina * 0.25;
    double scale = sqrt(fabs(c) / PI_D);
    double pang  = -(1.0 - a) * PI_D * 0.25;
    P[0] = Cchrp; P[1] = c;
    P[2] = cos(pang) * scale; P[3] = sin(pang) * scale;
    P[4] = c1; P[5] = c2; P[6] = c3;
  }
}

// ---------------------------------------------------------------------------
// K1: sinc table w[d] = (-1)^d/(pi*(d+1/2)); chirp/kernel tables with double
// range reduction of the quadratic phases.
// ---------------------------------------------------------------------------
__global__ void k_tables(const double* __restrict__ P,
                         float* __restrict__ wtab,
                         float* __restrict__ chre, float* __restrict__ chim,
                         float* __restrict__ kre,  float* __restrict__ kim) {
  const int i = blockIdx.x * blockDim.x + threadIdx.x;
  const double Cc = P[0], Ck = P[1];
  if (i < 2 * NSIG) {
    const int d = i - (NSIG - 1);
    wtab[i] = (float)(((d & 1) ? -1.0 : 1.0) / (PI_D * ((double)d + 0.5)));
  }
  if (i < 16384) {
    const long   n  = (long)i - (2 * NSIG - 2);
    const double th = fmod(Cc * (double)(n * n), 2.0 * PI_D);
    const bool   v  = i < (4 * NSIG - 3);
    chre[i] = v ? (float)cos(th) : 0.f;
    chim[i] = v ? (float)sin(th) : 0.f;
  }
  if (i < 32768) {
    const long   m  = (long)i - (4 * NSIG - 4);
    const double th = fmod(Ck * (double)(m * m), 2.0 * PI_D);
    const bool   v  = i < (8 * NSIG - 7);
    kre[i] = v ? (float)cos(th) : 0.f;
    kim[i] = v ? (float)sin(th) : 0.f;
  }
}

// ---------------------------------------------------------------------------
// K2: branch reduction (flip / centered DFT fallback; no-op for order=0.8).
// ---------------------------------------------------------------------------
__global__ void k_prepare(const float* __restrict__ x, const double* __restrict__ P,
                          float* __restrict__ fre, float* __restrict__ fim) {
  const int i = blockIdx.x * blockDim.x + threadIdx.x;
  if (i >= BATCH * NSIG) return;
  const int r = i >> 12, p = i & (NSIG - 1);
  const bool c1 = (P[4] != 0.0), c2 = (P[5] != 0.0), c3 = (P[6] != 0.0);
  const float* row = x + (size_t)r * NSIG;
  if (!c2 && !c3) {
    fre[i] = row[c1 ? (NSIG - 1 - p) : p];
    fim[i] = 0.f;
    return;
  }
  const double sgn = c2 ? -1.0 : 1.0;
  const double w0  = sgn * 2.0 * PI_D / (double)NSIG;
  const double pp  = (double)(p - NSIG / 2);
  double ar = 0.0, ai = 0.0;
  for (int q = 0; q < NSIG; ++q) {
    const double th = fmod(w0 * pp * (double)(q - NSIG / 2), 2.0 * PI_D);
    const double v  = (double)row[c1 ? (NSIG - 1 - q) : q];
    ar += v * cos(th); ai += v * sin(th);
  }
  const double s = 1.0 / sqrt((double)NSIG);
  fre[i] = (float)(ar * s); fim[i] = (float)(ai * s);
}

// ---------------------------------------------------------------------------
// K3: upsample GEMM: fup_odd[r,v] = sum_k fc[r,k] * w[v-k].
// A tile (16 rows, shared by all 8 waves) double-buffered in LDS via TDM.
// Each wave computes TWO 16-col tiles, reusing the A fragments.
// wtab window staged reversed in LDS (ascending Toeplitz reads).
// Dynamic LDS: 4*TILEF + WWIN floats (~34 KB).
// ---------------------------------------------------------------------------
__global__ __launch_bounds__(256)
void k_upsample_wmma(const float* __restrict__ fre, const float* __restrict__ fim,
                     const float* __restrict__ wtab,
                     float* __restrict__ ure, float* __restrict__ uim) {
  extern __shared__ float smem[];
  float* sA  = smem;                 // [buf][plane][TILEF]
  float* sWr = smem + 4 * TILEF;     // WWIN-1 used
  const int tid  = threadIdx.x;
  const int lane = tid & 31;
  const int wave = tid >> 5;
  const int colbase = blockIdx.x * WGCOLS;
  const int col0 = colbase + wave * 32;      // two tiles: col0, col0+16
  const int row0 = blockIdx.y * 16;
  const int m  = lane & 15;
  const int ks = (lane >> 4) * 2;
  const int n  = lane & 15;

  const float* Agr = fre + (size_t)row0 * NSIG;
  const float* Agi = fim + (size_t)row0 * NSIG;

  // sWr[j] = wtab[colbase + 4350 - j]; inner read idx = 255 - (dcol+n) + k
  for (int j = tid; j < WWIN - 1; j += 256) sWr[j] = wtab[colbase + 4350 - j];

  v8f aR0 = {0.f,0.f,0.f,0.f,0.f,0.f,0.f,0.f};
  v8f aI0 = {0.f,0.f,0.f,0.f,0.f,0.f,0.f,0.f};
  v8f aR1 = {0.f,0.f,0.f,0.f,0.f,0.f,0.f,0.f};
  v8f aI1 = {0.f,0.f,0.f,0.f,0.f,0.f,0.f,0.f};

#if USE_TDM
  if (wave == 0) {
    tdm_tile16x64(Agr, (unsigned)(uintptr_t)(sA + 0 * TILEF), NSIG);
    tdm_tile16x64(Agi, (unsigned)(uintptr_t)(sA + 1 * TILEF), NSIG);
    __builtin_amdgcn_s_wait_tensorcnt(0);
  }
  __syncthreads();
#else
  {
    float4 va, vb;
    stage_load(Agr, NSIG, 0, tid, &va);
    stage_load(Agi, NSIG, 0, tid, &vb);
    stage_store(sA + 0 * TILEF, tid, &va);
    stage_store(sA + 1 * TILEF, tid, &vb);
  }
  __syncthreads();
#endif

  int buf = 0;
  for (int kb = 0; kb < NSIG; kb += TCOLS) {
    const int nxt = kb + TCOLS;
#if USE_TDM
    if (nxt < NSIG && wave == 0) {
      tdm_tile16x64(Agr + nxt, (unsigned)(uintptr_t)(sA + (2 - 2 * buf) * TILEF), NSIG);
      tdm_tile16x64(Agi + nxt, (unsigned)(uintptr_t)(sA + (3 - 2 * buf) * TILEF), NSIG);
    }
#else
    float4 va, vb;
    const bool have = nxt < NSIG;
    if (have) {
      stage_load(Agr, NSIG, nxt, tid, &va);
      stage_load(Agi, NSIG, nxt, tid, &vb);
    }
#endif
    const float* pR = sA + (2 * buf + 0) * TILEF + m * TSTRIDE;
    const float* pI = sA + (2 * buf + 1) * TILEF + m * TSTRIDE;
    const int rb0 = 255 - (col0 - colbase) - n + kb;   // tile0 window base
#pragma unroll
    for (int ki = 0; ki < TCOLS; ki += 4) {
      const int kk = ki + ks;
      v2f ar; ar.x = pR[kk];              ar.y = pR[kk + 1];
      v2f ai; ai.x = pI[kk];              ai.y = pI[kk + 1];
      v2f b0; b0.x = sWr[rb0 + kk];       b0.y = sWr[rb0 + kk + 1];
      v2f b1; b1.x = sWr[rb0 - 16 + kk];  b1.y = sWr[rb0 - 16 + kk + 1];
      aR0 = wmma4(ar, b0, aR0);
      aI0 = wmma4(ai, b0, aI0);
      aR1 = wmma4(ar, b1, aR1);
      aI1 = wmma4(ai, b1, aI1);
    }
#if USE_TDM
    if (wave == 0) __builtin_amdgcn_s_wait_tensorcnt(0);
    __syncthreads();
#else
    __syncthreads();
    if (have) {
      stage_store(sA + (2 - 2 * buf) * TILEF, tid, &va);
      stage_store(sA + (3 - 2 * buf) * TILEF, tid, &vb);
    }
    __syncthreads();
#endif
    buf ^= 1;
  }

#pragma unroll
  for (int e = 0; e < 8; ++e) {
    const int mm = e + ((lane >> 4) * 8);
    const size_t o = (size_t)(row0 + mm) * NSIG + (col0 + n);
    ure[o]      = aR0[e];
    uim[o]      = aI0[e];
    ure[o + 16] = aR1[e];
    uim[o + 16] = aI1[e];
  }
}

// ---------------------------------------------------------------------------
// K4: assemble chirp-premultiplied interpolated signal g (K padded to 8192).
// ---------------------------------------------------------------------------
__global__ void k_assemble_g(const float* __restrict__ fre, const float* __restrict__ fim,
                             const float* __restrict__ ure, const float* __restrict__ uim,
                             const float* __restrict__ chre, const float* __restrict__ chim,
                             float* __restrict__ gre, float* __restrict__ gim) {
  const int i = blockIdx.x * blockDim.x + threadIdx.x;
  if (i >= BATCH * 2 * NSIG) return;
  const int r = i >> 13, u = i & (2 * NSIG - 1);
  if (u == 2 * NSIG - 1) { gre[i] = 0.f; gim[i] = 0.f; return; }
  const int v = u >> 1;
  const size_t s = (size_t)r * NSIG + v;
  const float vr = (u & 1) ? ure[s] : fre[s];
  const float vi = (u & 1) ? uim[s] : fim[s];
  const float cr = chre[NSIG - 1 + u], ci = chim[NSIG - 1 + u];
  gre[i] = vr * cr - vi * ci;
  gim[i] = vr * ci + vi * cr;
}

// ---------------------------------------------------------------------------
// K5: chirp-convolution GEMM (complex x complex, decimated outputs only):
//   Faf[r,t] = sum_u g[r,u] * kern[4N-4 + 2t - u]
// g tiles via TDM double buffer (row stride KDIM); per-WG kern window staged
// reversed in LDS.  Two 16-col tiles per wave reuse the A fragments.
// Epilogue fuses post-chirp + phase*scale -> interleaved complex64.
// Dynamic LDS: 4*TILEF + 2*KWIN floats (~87 KB; 320 KB WGP LDS -> 3 WGs).
// ---------------------------------------------------------------------------
__global__ __launch_bounds__(256)
void k_chirpconv_wmma(const float* __restrict__ gre, const float* __restrict__ gim,
                      const float* __restrict__ kre, const float* __restrict__ kim,
                      const float* __restrict__ chre, const float* __restrict__ chim,
                      const double* __restrict__ P, float* __restrict__ out) {
  extern __shared__ float smem[];
  float* sA   = smem;                    // [buf][plane][TILEF]
  float* sKre = smem + 4 * TILEF;        // KWIN-2 used
  float* sKim = sKre + KWIN;
  const int tid  = threadIdx.x;
  const int lane = tid & 31;
  const int wave = tid >> 5;
  const int tbase = blockIdx.x * WGCOLS;
  const int t0   = tbase + wave * 32;        // two tiles: t0, t0+16
  const int row0 = blockIdx.y * 16;
  const int m  = lane & 15;
  const int ks = (lane >> 4) * 2;
  const int n  = lane & 15;
  const int KDIM = 2 * NSIG;

  const float* Agr = gre + (size_t)row0 * KDIM;
  const float* Agi = gim + (size_t)row0 * KDIM;

  // sK*[j] = kern[kmax - j]; inner read idx = 510 - 2*(dt+n) + u
  const int kmax = 2 * tbase + 16890;
  for (int j = tid; j < KWIN - 2; j += 256) {
    sKre[j] = kre[kmax - j];
    sKim[j] = kim[kmax - j];
  }

  v8f Pa0 = {0.f,0.f,0.f,0.f,0.f,0.f,0.f,0.f};
  v8f Qa0 = {0.f,0.f,0.f,0.f,0.f,0.f,0.f,0.f};
  v8f Ra0 = {0.f,0.f,0.f,0.f,0.f,0.f,0.f,0.f};
  v8f Sa0 = {0.f,0.f,0.f,0.f,0.f,0.f,0.f,0.f};
  v8f Pa1 = {0.f,0.f,0.f,0.f,0.f,0.f,0.f,0.f};
  v8f Qa1 = {0.f,0.f,0.f,0.f,0.f,0.f,0.f,0.f};
  v8f Ra1 = {0.f,0.f,0.f,0.f,0.f,0.f,0.f,0.f};
  v8f Sa1 = {0.f,0.f,0.f,0.f,0.f,0.f,0.f,0.f};

#if USE_TDM
  if (wave == 0) {
    tdm_tile16x64(Agr, (unsigned)(uintptr_t)(sA + 0 * TILEF), KDIM);
    tdm_tile16x64(Agi, (unsigned)(uintptr_t)(sA + 1 * TILEF), KDIM);
    __builtin_amdgcn_s_wait_tensorcnt(0);
  }
  __syncthreads();
#else
  {
    float4 va, vb;
    stage_load(Agr, KDIM, 0, tid, &va);
    stage_load(Agi, KDIM, 0, tid, &vb);
    stage_store(sA + 0 * TILEF, tid, &va);
    stage_store(sA + 1 * TILEF, tid, &vb);
  }
  __syncthreads();
#endif

  int buf = 0;
  for (int kb = 0; kb < KDIM; kb += TCOLS) {
    const int nxt = kb + TCOLS;
#if USE_TDM
    if (nxt < KDIM && wave == 0) {
      tdm_tile16x64(Agr + nxt, (unsigned)(uintptr_t)(sA + (2 - 2 * buf) * TILEF), KDIM);
      tdm_tile16x64(Agi + nxt, (unsigned)(uintptr_t)(sA + (3 - 2 * buf) * TILEF), KDIM);
    }
#else
    float4 va, vb;
    const bool have = nxt < KDIM;
    if (have) {
      stage_load(Agr, KDIM, nxt, tid, &va);
      stage_load(Agi, KDIM, nxt, tid, &vb);
    }
#endif
    const float* pR = sA + (2 * buf + 0) * TILEF + m * TSTRIDE;
    const float* pI = sA + (2 * buf + 1) * TILEF + m * TSTRIDE;
    const int rb0 = 510 - 2 * ((t0 - tbase) + n) + kb;   // tile0; tile1 = rb0-32
#pragma unroll
    for (int ki = 0; ki < TCOLS; ki += 4) {
      const int kk = ki + ks;
      v2f ar; ar.x = pR[kk];                 ar.y = pR[kk + 1];
      v2f ai; ai.x = pI[kk];                 ai.y = pI[kk + 1];
      v2f br0; br0.x = sKre[rb0 + kk];       br0.y = sKre[rb0 + kk + 1];
      v2f bi0; bi0.x = sKim[rb0 + kk];       bi0.y = sKim[rb0 + kk + 1];
      v2f br1; br1.x = sKre[rb0 - 32 + kk];  br1.y = sKre[rb0 - 32 + kk + 1];
      v2f bi1; bi1.x = sKim[rb0 - 32 + kk];  bi1.y = sKim[rb0 - 32 + kk + 1];
      Pa0 = wmma4(ar, br0, Pa0);
      Qa0 = wmma4(ai, bi0, Qa0);
      Ra0 = wmma4(ar, bi0, Ra0);
      Sa0 = wmma4(ai, br0, Sa0);
      Pa1 = wmma4(ar, br1, Pa1);
      Qa1 = wmma4(ai, bi1, Qa1);
      Ra1 = wmma4(ar, bi1, Ra1);
      Sa1 = wmma4(ai, br1, Sa1);
    }
#if USE_TDM
    if (wave == 0) __builtin_amdgcn_s_wait_tensorcnt(0);
    __syncthreads();
#else
    __syncthreads();
    if (have) {
      stage_store(sA + (2 - 2 * buf) * TILEF, tid, &va);
      stage_store(sA + (3 - 2 * buf) * TILEF, tid, &vb);
    }
    __syncthreads();
#endif
    buf ^= 1;
  }

  const float pfr = (float)P[2], pfi = (float)P[3];
#pragma unroll
  for (int e = 0; e < 8; ++e) {
    const int mm = e + ((lane >> 4) * 8);
#pragma unroll
    for (int tt = 0; tt < 2; ++tt) {
      const int t = t0 + 16 * tt + n;
      const float cr = (tt ? Pa1[e] : Pa0[e]) - (tt ? Qa1[e] : Qa0[e]);
      const float ci = (tt ? Ra1[e] : Ra0[e]) + (tt ? Sa1[e] : Sa0[e]);
      const float hr = chre[NSIG - 1 + 2 * t], hi = chim[NSIG - 1 + 2 * t];
      const float m1r = cr * hr - ci * hi;
      const float m1i = cr * hi + ci * hr;
      const size_t o = ((size_t)(row0 + mm) * NSIG + t) * 2;
      out[o + 0] = m1r * pfr - m1i * pfi;
      out[o + 1] = m1r * pfi + m1i * pfr;
    }
  }
}

// ---------------------------------------------------------------------------
extern "C" void kernel_launch(void* const* d_in, const int* in_sizes, int n_in,
                              void* d_out, int out_size, void* d_ws, size_t ws_size,
                              hipStream_t stream) {
  (void)in_sizes; (void)n_in; (void)out_size; (void)ws_size;
  const float* x     = (const float*)d_in[0];
  const float* order = (const float*)d_in[1];
  float*  ws = (float*)d_ws;
  double* P  = (double*)d_ws;

  float* wtab = ws + OFF_WTAB;
  float* chre = ws + OFF_CHRP_RE; float* chim = ws + OFF_CHRP_IM;
  float* kre  = ws + OFF_KERN_RE; float* kim  = ws + OFF_KERN_IM;
  float* fre  = ws + OFF_FC_RE;   float* fim  = ws + OFF_FC_IM;
  float* ure  = ws + OFF_UP_RE;   float* uim  = ws + OFF_UP_IM;
  float* gre  = ws + OFF_G_RE;    float* gim  = ws + OFF_G_IM;

  const size_t lds1 = (4 * TILEF + WWIN) * sizeof(float);            // ~34 KB
  const size_t lds2 = (4 * TILEF + 2 * KWIN) * sizeof(float);        // ~87 KB

  k_setup<<<1, 32, 0, stream>>>(order, P);
  k_tables<<<32768 / 256, 256, 0, stream>>>(P, wtab, chre, chim, kre, kim);
  k_prepare<<<(BATCH * NSIG) / 256, 256, 0, stream>>>(x, P, fre, fim);
  k_upsample_wmma<<<dim3(NSIG / WGCOLS, BATCH / 16), 256, lds1, stream>>>(
      fre, fim, wtab, ure, uim);
  k_assemble_g<<<(BATCH * 2 * NSIG) / 256, 256, 0, stream>>>(
      fre, fim, ure, uim, chre, chim, gre, gim);
  k_chirpconv_wmma<<<dim3(NSIG / WGCOLS, BATCH / 16), 256, lds2, stream>>>(
      gre, gim, kre, kim, chre, chim, P, (float*)d_out);
}